// Net_69707319214346
// MI455X (gfx1250) — compile-verified
//
#include <hip/hip_runtime.h>

// GIN-style GNN forward for MI455X (gfx1250, wave32).
// Bandwidth-bound (edge scatter ~1.2GB @ 23.3TB/s; h (12.8MB) is L2-resident);
// GEMMs (~3 GFLOP) use full-precision V_WMMA_F32_16X16X4_F32 to keep fp32
// reference semantics. Weights staged in LDS as interleaved K-pairs so every
// WMMA B operand is one aligned ds_load_b64 (no repacking movs).

#define HID 64
#define BN_EPS 1e-5f
#define STAT_BLOCKS 256

typedef __attribute__((ext_vector_type(2))) float v2f;
typedef __attribute__((ext_vector_type(8))) float v8f;

// ---------------- utility / elementwise kernels ----------------

__global__ void k_fill(float* __restrict__ p, float v, int n) {
  int i = blockIdx.x * blockDim.x + threadIdx.x;
  if (i < n) p[i] = v;
}

// out[node,c] = x[node,0]*W[0,c] + x[node,1]*W[1,c] + b[c]   (IN_C == 2)
__global__ void k_lin_in(const float* __restrict__ x, const float* __restrict__ W,
                         const float* __restrict__ b, float* __restrict__ out, int n) {
  int i = blockIdx.x * blockDim.x + threadIdx.x;
  if (i >= n * HID) return;
  int node = i >> 6, c = i & (HID - 1);
  float x0 = x[node * 2 + 0], x1 = x[node * 2 + 1];
  out[i] = fmaf(x0, W[c], fmaf(x1, W[HID + c], b[c]));
}

// Deterministic per-column sum / sum-of-squares partials (fixed reduction order).
__global__ void k_colstats(const float* __restrict__ a, float* __restrict__ partials,
                           int nrows) {
  __shared__ float s[4][2 * HID];
  int tid = threadIdx.x;            // 256 threads
  int c   = tid & (HID - 1);
  int sub = tid >> 6;               // 0..3
  float sum = 0.f, sq = 0.f;
  for (int r = blockIdx.x * 4 + sub; r < nrows; r += gridDim.x * 4) {
    float v = a[r * HID + c];
    sum += v; sq += v * v;
  }
  s[sub][c] = sum; s[sub][HID + c] = sq;
  __syncthreads();
  if (tid < 2 * HID) {
    float t = s[0][tid] + s[1][tid] + s[2][tid] + s[3][tid];
    partials[blockIdx.x * 2 * HID + tid] = t;
  }
}

// scale/shift per channel: y = g*(x-mu)*rsqrt(var+eps)+b  ==  x*sc + sh
__global__ void k_bnfinal(const float* __restrict__ partials, int nblocks,
                          const float* __restrict__ g, const float* __restrict__ b,
                          float* __restrict__ ss, int nrows) {
  int c = threadIdx.x;
  if (c >= HID) return;
  float sum = 0.f, sq = 0.f;
  for (int k = 0; k < nblocks; k++) {
    sum += partials[k * 2 * HID + c];
    sq  += partials[k * 2 * HID + HID + c];
  }
  float inv_n = 1.0f / (float)nrows;
  float mu  = sum * inv_n;
  float var = sq * inv_n - mu * mu;
  float inv = rsqrtf(var + BN_EPS);
  float sc  = g[c] * inv;
  ss[c]       = sc;
  ss[HID + c] = b[c] - sc * mu;
}

// h = relu(a*sc + sh); optionally pool += h (node_pool accumulation)
__global__ void k_bn_relu(const float* __restrict__ a, const float* __restrict__ ss,
                          float* __restrict__ h, float* __restrict__ pool, int n) {
  int i = blockIdx.x * blockDim.x + threadIdx.x;
  if (i >= n * HID) return;
  int c = i & (HID - 1);
  float v = fmaxf(fmaf(a[i], ss[c], ss[HID + c]), 0.0f);
  h[i] = v;
  if (pool) pool[i] += v;
}

__global__ void k_count(const int* __restrict__ idx, float* __restrict__ cnt, int n) {
  int i = blockIdx.x * blockDim.x + threadIdx.x;
  if (i < n) atomicAdd(&cnt[idx[i]], 1.0f);
}

__global__ void k_recip_clamp(float* __restrict__ p, int n) {
  int i = blockIdx.x * blockDim.x + threadIdx.x;
  if (i < n) p[i] = 1.0f / fmaxf(p[i], 1.0f);
}

// agg[dst[e], :] += h[src[e], :]   (16 threads/edge, float4 gather, f32 atomics to L2)
__global__ void k_scatter(const int* __restrict__ src, const int* __restrict__ dst,
                          const float* __restrict__ h, float* __restrict__ agg, int E) {
  int i = blockIdx.x * blockDim.x + threadIdx.x;
  if (i >= E * 16) return;
  int e = i >> 4, q = (i & 15) * 4;
  int s = src[e], d = dst[e];
  const float4 v = *(const float4*)(h + (size_t)s * HID + q);
  float* p = agg + (size_t)d * HID + q;
  atomicAdd(p + 0, v.x); atomicAdd(p + 1, v.y);
  atomicAdd(p + 2, v.z); atomicAdd(p + 3, v.w);
}

// out = h + agg * invdeg[row]
__global__ void k_combine(const float* __restrict__ h, const float* __restrict__ agg,
                          const float* __restrict__ invdeg, float* __restrict__ out, int n) {
  int i = blockIdx.x * blockDim.x + threadIdx.x;
  if (i >= n * HID) return;
  int r = i >> 6;
  out[i] = fmaf(agg[i], invdeg[r], h[i]);
}

// gsum[batch[node], :] += h[node, :]
__global__ void k_gscatter(const int* __restrict__ batch, const float* __restrict__ h,
                           float* __restrict__ gsum, int n) {
  int i = blockIdx.x * blockDim.x + threadIdx.x;
  if (i >= n * 16) return;
  int node = i >> 4, q = (i & 15) * 4;
  int g = batch[node];
  const float4 v = *(const float4*)(h + (size_t)node * HID + q);
  float* p = gsum + (size_t)g * HID + q;
  atomicAdd(p + 0, v.x); atomicAdd(p + 1, v.y);
  atomicAdd(p + 2, v.z); atomicAdd(p + 3, v.w);
}

// ---------------- WMMA fp32 GEMM: out[nrows,64] = (A*rowscale) @ W + bias ----------------
// flags bit0: relu epilogue; bit1: accumulate into out (out += result).
// 128 threads = 4 wave32 waves; one 16-row M tile per wave, 4 16x16 N-tiles each, K=64 in 16 steps of 4.
// A 16x4 layout (ISA 7.12.2): lane l holds row M=l&15; VGPR0/1 = K{0,1} (l<16) or K{2,3}.
// B 4x16 mirrors C/D striping: VGPR0 = K0 (lanes0-15) / K2 (lanes16-31), VGPR1 = K1/K3.
// LDS holds W interleaved by K-pairs: Wp[k2*64+col] = {W[2k2][col], W[2k2+1][col]},
// so each B operand is a single 8-byte-aligned ds_load_b64 into an even VGPR pair.
// C/D: VGPR r, lane l -> row r + 8*(l>>4), col l&15.
__global__ void __launch_bounds__(128)
k_gemm64(const float* __restrict__ A, const float* __restrict__ W,
         const float* __restrict__ bias, const float* __restrict__ rowscale,
         float* __restrict__ out, int nrows, int flags) {
  __shared__ v2f  Wp[(HID / 2) * HID];   // 32 k-pairs x 64 cols = 16 KB
  __shared__ float bl[HID];
  int tid = threadIdx.x;
  for (int i = tid; i < (HID / 2) * HID; i += 128) {
    int k2 = i >> 6, col = i & (HID - 1);
    v2f t;
    t.x = W[(2 * k2) * HID + col];
    t.y = W[(2 * k2 + 1) * HID + col];
    Wp[i] = t;
  }
  if (tid < HID) bl[tid] = bias[tid];
  __syncthreads();

  const int wave = tid >> 5;
  const int lane = tid & 31;
  const int half = lane >> 4;        // selects K-pair for A/B operands
  const int m    = lane & 15;
  const int tileRow0 = (blockIdx.x * 4 + wave) * 16;

  const int  arow_i  = tileRow0 + m;
  const bool avalid  = arow_i < nrows;
  const float rs     = avalid ? (rowscale ? rowscale[arow_i] : 1.0f) : 0.0f;
  const float* arow  = A + (size_t)(avalid ? arow_i : 0) * HID;

  v8f acc0 = {}, acc1 = {}, acc2 = {}, acc3 = {};
#pragma unroll
  for (int ks = 0; ks < 16; ks++) {
    const int k0 = ks * 4 + half * 2;      // first K of this lane-half's pair
    const int k2 = ks * 2 + half;          // K-pair index into Wp
    v2f a;
    a.x = arow[k0] * rs;                   // rs==0 zeroes padded rows; no EXEC masking
    a.y = arow[k0 + 1] * rs;
    const v2f* bp = &Wp[k2 * HID + m];
    v2f b0 = bp[0 * 16];
    v2f b1 = bp[1 * 16];
    v2f b2 = bp[2 * 16];
    v2f b3 = bp[3 * 16];
    acc0 = __builtin_amdgcn_wmma_f32_16x16x4_f32(false, a, false, b0, (short)0, acc0, false, false);
    acc1 = __builtin_amdgcn_wmma_f32_16x16x4_f32(false, a, false, b1, (short)0, acc1, false, false);
    acc2 = __builtin_amdgcn_wmma_f32_16x16x4_f32(false, a, false, b2, (short)0, acc2, false, false);
    acc3 = __builtin_amdgcn_wmma_f32_16x16x4_f32(false, a, false, b3, (short)0, acc3, false, false);
  }

  const bool relu  = flags & 1;
  const bool accum = flags & 2;
#pragma unroll
  for (int r = 0; r < 8; r++) {
    int orow = tileRow0 + half * 8 + r;
    if (orow >= nrows) continue;
    float* op = out + (size_t)orow * HID;
#pragma unroll
    for (int nt = 0; nt < 4; nt++) {
      int col = nt * 16 + m;
      float v = (nt == 0 ? acc0[r] : nt == 1 ? acc1[r] : nt == 2 ? acc2[r] : acc3[r]) + bl[col];
      if (relu) v = fmaxf(v, 0.0f);
      if (accum) op[col] += v; else op[col] = v;
    }
  }
}

// ---------------- host-side orchestration ----------------

extern "C" void kernel_launch(void* const* d_in, const int* in_sizes, int n_in,
                              void* d_out, int out_size, void* d_ws, size_t ws_size,
                              hipStream_t stream) {
  (void)n_in; (void)out_size; (void)ws_size;
  const int N = in_sizes[0] / 2;     // x is [N,2]
  const int E = in_sizes[1] / 2;     // edge_index is [2,E]
  const int G = 512;

  const float* x     = (const float*)d_in[0];
  const int*   ei    = (const int*)d_in[1];
  const int*   src   = ei;
  const int*   dst   = ei + E;
  const int*   batch = (const int*)d_in[2];
  // params flattened in setup_inputs() insertion order
  const float* lin11W = (const float*)d_in[3];
  const float* lin11b = (const float*)d_in[4];
  const float* bn11g  = (const float*)d_in[5];
  const float* bn11b  = (const float*)d_in[6];
  const float* lin12W = (const float*)d_in[7];
  const float* lin12b = (const float*)d_in[8];
  const float *convW1[3], *convb1[3], *convW2[3], *convb2[3];
  for (int i = 0; i < 3; i++) {
    convW1[i] = (const float*)d_in[9 + 4 * i];
    convb1[i] = (const float*)d_in[10 + 4 * i];
    convW2[i] = (const float*)d_in[11 + 4 * i];
    convb2[i] = (const float*)d_in[12 + 4 * i];
  }
  const float *bng[3], *bnb[3];
  for (int i = 0; i < 3; i++) {
    bng[i] = (const float*)d_in[21 + 2 * i];
    bnb[i] = (const float*)d_in[22 + 2 * i];
  }
  const float *predW[4], *predb[4];
  for (int i = 0; i < 4; i++) {
    predW[i] = (const float*)d_in[27 + 2 * i];
    predb[i] = (const float*)d_in[28 + 2 * i];
  }

  // workspace carve-up (all fp32): ~51.5 MB total
  float* w = (float*)d_ws;
  const size_t NH = (size_t)N * HID;
  float* h      = w; w += NH;
  float* tA     = w; w += NH;
  float* tB     = w; w += NH;
  float* agg    = w; w += NH;
  float* invdeg = w; w += N;
  float* invcnt = w; w += G;
  float* gsum   = w; w += (size_t)G * HID;
  float* ss     = w; w += 2 * HID;
  float* partials = w; w += 2 * HID * STAT_BLOCKS;

  float* out_node = (float*)d_out;        // [N,64] node pool (sum of conv h's)
  float* out_g    = (float*)d_out + NH;   // [G,64] graph-pool head sum

  const int T = 256;
  auto cdiv = [](long a, long b) { return (int)((a + b - 1) / b); };
  const int nNH  = (int)NH;
  const int gNH  = cdiv(nNH, T);
  const int gSc  = cdiv((long)E * 16, T);
  const int gGs  = cdiv((long)N * 16, T);
  const int gGemmN = cdiv(N, 64);
  const int gGemmG = cdiv(G, 64);

  // zero accumulators
  k_fill<<<gNH, T, 0, stream>>>(out_node, 0.0f, nNH);
  k_fill<<<cdiv(G * HID, T), T, 0, stream>>>(out_g, 0.0f, G * HID);
  k_fill<<<cdiv(N, T), T, 0, stream>>>(invdeg, 0.0f, N);
  k_fill<<<cdiv(G, T), T, 0, stream>>>(invcnt, 0.0f, G);

  // degrees + graph counts -> reciprocals
  k_count<<<cdiv(E, T), T, 0, stream>>>(dst, invdeg, E);
  k_recip_clamp<<<cdiv(N, T), T, 0, stream>>>(invdeg, N);
  k_count<<<cdiv(N, T), T, 0, stream>>>(batch, invcnt, N);
  k_recip_clamp<<<cdiv(G, T), T, 0, stream>>>(invcnt, G);

  // h0 = lin1_2(relu(BN(lin1_1(x))))
  k_lin_in<<<gNH, T, 0, stream>>>(x, lin11W, lin11b, tA, N);
  k_colstats<<<STAT_BLOCKS, T, 0, stream>>>(tA, partials, N);
  k_bnfinal<<<1, HID, 0, stream>>>(partials, STAT_BLOCKS, bn11g, bn11b, ss, N);
  k_bn_relu<<<gNH, T, 0, stream>>>(tA, ss, tB, nullptr, N);
  k_gemm64<<<gGemmN, 128, 0, stream>>>(tB, lin12W, lin12b, nullptr, h, N, 0);

  // graph-pool head 0 on h0
  k_fill<<<cdiv(G * HID, T), T, 0, stream>>>(gsum, 0.0f, G * HID);
  k_gscatter<<<gGs, T, 0, stream>>>(batch, h, gsum, N);
  k_gemm64<<<gGemmG, 128, 0, stream>>>(gsum, predW[0], predb[0], invcnt, out_g, G, 2);

  for (int L = 0; L < 3; L++) {
    // mean aggregation over in-neighbors
    k_fill<<<gNH, T, 0, stream>>>(agg, 0.0f, nNH);
    k_scatter<<<gSc, T, 0, stream>>>(src, dst, h, agg, E);
    k_combine<<<gNH, T, 0, stream>>>(h, agg, invdeg, tA, N);
    // MLP: relu(z @ l1 + b1) @ l2 + b2
    k_gemm64<<<gGemmN, 128, 0, stream>>>(tA, convW1[L], convb1[L], nullptr, tB, N, 1);
    k_gemm64<<<gGemmN, 128, 0, stream>>>(tB, convW2[L], convb2[L], nullptr, tA, N, 0);
    // BN + relu, accumulate node pool
    k_colstats<<<STAT_BLOCKS, T, 0, stream>>>(tA, partials, N);
    k_bnfinal<<<1, HID, 0, stream>>>(partials, STAT_BLOCKS, bng[L], bnb[L], ss, N);
    k_bn_relu<<<gNH, T, 0, stream>>>(tA, ss, h, out_node, N);
    // graph-pool head L+1
    k_fill<<<cdiv(G * HID, T), T, 0, stream>>>(gsum, 0.0f, G * HID);
    k_gscatter<<<gGs, T, 0, stream>>>(batch, h, gsum, N);
    k_gemm64<<<gGemmG, 128, 0, stream>>>(gsum, predW[L + 1], predb[L + 1], invcnt, out_g, G, 2);
  }
}